// MambaTemporal_82214263980612
// MI455X (gfx1250) — compile-verified
//
#include <hip/hip_runtime.h>
#include <hip/hip_bf16.h>

// ---------------- problem constants (from reference) ----------------
#define BB      16
#define TT      96
#define NN      325
#define DM      64
#define DSTATE  16
#define DCONV   4
#define DI      128            // expand * Dm
#define DTR     4              // Dm / 16
#define NSEQ    (BB * NN)      // 5200 independent sequences
#define WIN_N   (2 * DI)       // 256
#define WX_N    (DTR + 2 * DSTATE) // 36

// LDS layout (floats)
#define OFF_XS   0                         // (96 x 64)   xs tile
#define OFF_XI   (OFF_XS + TT * DM)        // (96 x 128)  xi, later reused as y
#define OFF_XC   (OFF_XI + TT * DI)        // (96 x 128)  xc, later reused as g
#define OFF_DBL  (OFF_XC + TT * DI)        // (96 x 36)   dt_r | B | C
#define SMEM_FLOATS (OFF_DBL + TT * WX_N)  // 34176 floats = 136704 bytes

typedef float v2f __attribute__((ext_vector_type(2)));
typedef float v8f __attribute__((ext_vector_type(8)));

__device__ __forceinline__ v8f wmma_f32(v2f a, v2f b, v8f c) {
    // D = A(16x4) * B(4x16) + C(16x16), all f32, wave32
    return __builtin_amdgcn_wmma_f32_16x16x4_f32(
        false, a, false, b, (short)0, c, false, false);
}

__device__ __forceinline__ float silu_f(float x) {
    return x / (1.0f + __expf(-x));
}
__device__ __forceinline__ float softplus_f(float x) {
    return (x > 20.0f) ? x : log1pf(__expf(x));
}

// A fragment: lane m = lane&15 holds row m; lanes[16:31] hold K+2..K+3
__device__ __forceinline__ v2f frag_a_lds(const float* buf, int ld, int mbase,
                                          int k, int m, int kb) {
    v2f a;
    a.x = buf[(mbase + m) * ld + k + kb];
    a.y = buf[(mbase + m) * ld + k + kb + 1];
    return a;
}
// B fragment: lane n = lane&15 holds col n; lanes[16:31] hold K+2..K+3
__device__ __forceinline__ v2f frag_b_glb(const float* W, int ldn, int nbase,
                                          int k, int n, int kb) {
    v2f b;
    b.x = W[(k + kb) * ldn + nbase + n];
    b.y = W[(k + kb + 1) * ldn + nbase + n];
    return b;
}

__global__ void __launch_bounds__(256)
mamba_fused_kernel(const float* __restrict__ x,       // (B,T,N,Dm)
                   const float* __restrict__ W_in,    // (Dm, 2*di)
                   const float* __restrict__ conv_w,  // (di, 4)
                   const float* __restrict__ conv_b,  // (di)
                   const float* __restrict__ W_x,     // (di, 36)
                   const float* __restrict__ W_dt,    // (4, di)
                   const float* __restrict__ b_dt,    // (di)
                   const float* __restrict__ A_log,   // (di, 16)
                   const float* __restrict__ D_skip,  // (di)
                   const float* __restrict__ W_out,   // (di, Dm)
                   float* __restrict__ out)           // (B,T,N,Dm)
{
    extern __shared__ float smem[];
    float* s_xs  = smem + OFF_XS;   // ld = DM
    float* s_xi  = smem + OFF_XI;   // ld = DI   (reused as y after the scan)
    float* s_xc  = smem + OFF_XC;   // ld = DI   (reused as g after gating)
    float* s_dbl = smem + OFF_DBL;  // ld = WX_N

    const int seq  = blockIdx.x;           // 0..5199
    const int bidx = seq / NN;
    const int nidx = seq % NN;
    const int tid  = threadIdx.x;
    const int lane = tid & 31;
    const int wave = tid >> 5;             // 0..7
    const int m16  = lane & 15;            // row (A) / col (B) within tile
    const int hi   = lane >> 4;            // 0 or 1
    const int kb   = hi << 1;              // K sub-offset 0 or 2

    // ---------------- Stage A: load xs tile (96 x 64) ----------------
    // xs[t][dm] = x[b][t][n][dm]; rows of 64 floats are contiguous.
    for (int i = tid; i < TT * (DM / 4); i += 256) {
        int t = i >> 4;           // / 16
        int q = i & 15;           // float4 within row
        const float4 v = *reinterpret_cast<const float4*>(
            x + (((size_t)bidx * TT + t) * NN + nidx) * DM + q * 4);
        *reinterpret_cast<float4*>(s_xs + t * DM + q * 4) = v;
    }
    __syncthreads();

    // ---------------- Stage B: xi = xs @ W_in[:, 0:128] ----------------
    // 6 M-tiles x 8 N-tiles = 48 tiles over 8 waves
    for (int tile = wave; tile < 6 * 8; tile += 8) {
        const int tm = tile >> 3, tn = tile & 7;
        v8f acc = {};
        #pragma unroll
        for (int k = 0; k < DM; k += 4) {
            v2f a = frag_a_lds(s_xs, DM, tm * 16, k, m16, kb);
            v2f b = frag_b_glb(W_in, WIN_N, tn * 16, k, m16, kb);
            acc = wmma_f32(a, b, acc);
        }
        #pragma unroll
        for (int r = 0; r < 8; ++r)
            s_xi[(tm * 16 + r + 8 * hi) * DI + tn * 16 + m16] = acc[r];
    }
    __syncthreads();

    // ---------------- Stage C: depthwise causal conv + bias + SiLU ------
    if (tid < DI) {
        const int d = tid;
        const float w0 = conv_w[d * 4 + 0], w1 = conv_w[d * 4 + 1];
        const float w2 = conv_w[d * 4 + 2], w3 = conv_w[d * 4 + 3];
        const float cb = conv_b[d];
        float xm3 = 0.f, xm2 = 0.f, xm1 = 0.f;
        for (int t = 0; t < TT; ++t) {
            const float cur = s_xi[t * DI + d];
            const float v = w0 * xm3 + w1 * xm2 + w2 * xm1 + w3 * cur + cb;
            s_xc[t * DI + d] = silu_f(v);
            xm3 = xm2; xm2 = xm1; xm1 = cur;
        }
    }
    __syncthreads();

    // ---------------- Stage D: dbl = xc @ W_x  (96 x 36, pad N to 48) ---
    for (int tile = wave; tile < 6 * 3; tile += 8) {
        const int tm = tile / 3, tn = tile % 3;
        const int col = tn * 16 + m16;
        const bool ok = (col < WX_N);
        v8f acc = {};
        #pragma unroll 4
        for (int k = 0; k < DI; k += 4) {
            v2f a = frag_a_lds(s_xc, DI, tm * 16, k, m16, kb);
            v2f b;
            b.x = ok ? W_x[(k + kb) * WX_N + col] : 0.0f;
            b.y = ok ? W_x[(k + kb + 1) * WX_N + col] : 0.0f;
            acc = wmma_f32(a, b, acc);
        }
        if (ok) {
            #pragma unroll
            for (int r = 0; r < 8; ++r)
                s_dbl[(tm * 16 + r + 8 * hi) * WX_N + col] = acc[r];
        }
    }
    __syncthreads();

    // ---------------- Stage E: selective-scan (y overwrites s_xi) -------
    {
        const int d     = tid >> 1;        // channel, 2 threads per channel
        const int shalf = tid & 1;         // which 8 of the 16 states
        const int sbase = shalf * 8;

        float Av[8];
        #pragma unroll
        for (int i = 0; i < 8; ++i) Av[i] = -__expf(A_log[d * DSTATE + sbase + i]);
        float wdt0 = W_dt[0 * DI + d], wdt1 = W_dt[1 * DI + d];
        float wdt2 = W_dt[2 * DI + d], wdt3 = W_dt[3 * DI + d];
        const float bdt = b_dt[d];
        const float dsk = D_skip[d];
        float h[8];
        #pragma unroll
        for (int i = 0; i < 8; ++i) h[i] = 0.f;

        for (int t = 0; t < TT; ++t) {
            const float xt  = s_xc[t * DI + d];
            const float* rw = s_dbl + t * WX_N;
            const float pre = rw[0] * wdt0 + rw[1] * wdt1 +
                              rw[2] * wdt2 + rw[3] * wdt3 + bdt;
            const float dt = softplus_f(pre);
            float acc = 0.f;
            #pragma unroll
            for (int i = 0; i < 8; ++i) {
                const float Bv = rw[DTR + sbase + i];
                const float Cv = rw[DTR + DSTATE + sbase + i];
                const float dA = __expf(dt * Av[i]);
                h[i] = dA * h[i] + dt * Bv * xt;
                acc += h[i] * Cv;
            }
            acc += __shfl_xor(acc, 1, 32);           // combine state halves
            if (shalf == 0) s_xi[t * DI + d] = acc + xt * dsk;  // y
        }
    }
    __syncthreads();

    // ---------------- Stage F: g = y * silu(z), z = xs @ W_in[:,128:] ---
    // g overwrites s_xc. 48 tiles over 8 waves.
    for (int tile = wave; tile < 6 * 8; tile += 8) {
        const int tm = tile >> 3, tn = tile & 7;
        v8f acc = {};
        #pragma unroll
        for (int k = 0; k < DM; k += 4) {
            v2f a = frag_a_lds(s_xs, DM, tm * 16, k, m16, kb);
            v2f b = frag_b_glb(W_in, WIN_N, DI + tn * 16, k, m16, kb);
            acc = wmma_f32(a, b, acc);
        }
        #pragma unroll
        for (int r = 0; r < 8; ++r) {
            const int t = tm * 16 + r + 8 * hi;
            const int j = tn * 16 + m16;
            const float zv = acc[r];
            s_xc[t * DI + j] = s_xi[t * DI + j] * silu_f(zv);
        }
    }
    __syncthreads();

    // ---------------- Stage G: out = g @ W_out  (96 x 64) ---------------
    for (int tile = wave; tile < 6 * 4; tile += 8) {
        const int tm = tile >> 2, tn = tile & 3;
        v8f acc = {};
        #pragma unroll 4
        for (int k = 0; k < DI; k += 4) {
            v2f a = frag_a_lds(s_xc, DI, tm * 16, k, m16, kb);
            v2f b = frag_b_glb(W_out, DM, tn * 16, k, m16, kb);
            acc = wmma_f32(a, b, acc);
        }
        #pragma unroll
        for (int r = 0; r < 8; ++r) {
            const int t  = tm * 16 + r + 8 * hi;
            const int dm = tn * 16 + m16;
            out[(((size_t)bidx * TT + t) * NN + nidx) * DM + dm] = acc[r];
        }
    }
}

extern "C" void kernel_launch(void* const* d_in, const int* in_sizes, int n_in,
                              void* d_out, int out_size, void* d_ws, size_t ws_size,
                              hipStream_t stream) {
    (void)in_sizes; (void)n_in; (void)out_size; (void)d_ws; (void)ws_size;
    const float* x      = (const float*)d_in[0];
    const float* W_in   = (const float*)d_in[1];
    const float* conv_w = (const float*)d_in[2];
    const float* conv_b = (const float*)d_in[3];
    const float* W_x    = (const float*)d_in[4];
    const float* W_dt   = (const float*)d_in[5];
    const float* b_dt   = (const float*)d_in[6];
    const float* A_log  = (const float*)d_in[7];
    const float* D_skip = (const float*)d_in[8];
    const float* W_out  = (const float*)d_in[9];
    float* out = (float*)d_out;

    const size_t smem_bytes = SMEM_FLOATS * sizeof(float); // ~134 KB of 320 KB/WGP
    mamba_fused_kernel<<<NSEQ, 256, smem_bytes, stream>>>(
        x, W_in, conv_w, conv_b, W_x, W_dt, b_dt, A_log, D_skip, W_out, out);
}